// MultiHashRouterFast_40664750359235
// MI455X (gfx1250) — compile-verified
//
#include <hip/hip_runtime.h>
#include <hip/hip_bf16.h>

// MultiHashRouter for MI455X (gfx1250, wave32).
// One wave per row: lane l handles columns 2l, 2l+1 of the 64-col prefix.
// Row slice (256B) is staged global->LDS with the CDNA5 async-copy path
// (global_load_async_to_lds_b64 SADDR form, ASYNCcnt + s_wait_asynccnt),
// then reduced with a wave32 XOR butterfly.

constexpr int DIM  = 4096;   // row stride of hidden_states
constexpr int DLIM = 64;     // columns actually used
constexpr int WAVES_PER_BLOCK = 8;

__device__ __forceinline__ unsigned hash_term(float x, int col /*0-based*/) {
    // sign in {-1,0,1}; (sign & 3): -1 -> 3, 0 -> 0, 1 -> 1
    int s = (x > 0.0f) - (x < 0.0f);
    int b = (int)fabsf(x);            // truncate toward zero
    b = b > 7 ? 7 : b;                // clip to [0,7] (b >= 0 already)
    unsigned dv = ((unsigned)(s & 3) << 2) | (unsigned)(b & 7);
    return dv * (unsigned)(col + 1);  // term, XOR-reduced across columns
}

__global__ __launch_bounds__(256) void
router_kernel(const float* __restrict__ hs,
              float* __restrict__ sel_out,    // [n,2] as float
              float* __restrict__ w_out,      // [n,2]
              float* __restrict__ mask_out,   // [n,64]
              int nrows) {
    __shared__ float stage[WAVES_PER_BLOCK * DLIM];   // 2 KB

    const int lane = (int)(threadIdx.x & 31u);
    const int wave = (int)(threadIdx.x >> 5u);
    const int row  = (int)blockIdx.x * WAVES_PER_BLOCK + wave;
    if (row >= nrows) return;   // wave-uniform

    // ---- async stage: each lane DMAs 8 bytes (2 floats) global -> LDS ----
    // GVS addressing: 64-bit SGPR base + 32-bit per-lane byte offset
    // (max offset 16384*4096*4 = 2^28 bytes, fits comfortably in 32 bits).
    unsigned gbyte  = ((unsigned)row * (unsigned)DIM + (unsigned)(lane * 2)) * 4u;
    unsigned ldsOff = (unsigned)(unsigned long long)(const void*)&stage[wave * DLIM + lane * 2];
    asm volatile("global_load_async_to_lds_b64 %0, %1, %2"
                 :: "v"(ldsOff), "v"(gbyte), "s"(hs)
                 : "memory");
#if __has_builtin(__builtin_amdgcn_s_wait_asynccnt)
    __builtin_amdgcn_s_wait_asynccnt(0);
#else
    asm volatile("s_wait_asynccnt 0x0" ::: "memory");
#endif
    float2 v = *(const float2*)&stage[wave * DLIM + lane * 2];

    // ---- per-lane partial hash, then wave32 XOR butterfly ----
    unsigned part = hash_term(v.x, lane * 2) ^ hash_term(v.y, lane * 2 + 1);
    #pragma unroll
    for (int off = 16; off > 0; off >>= 1)
        part ^= (unsigned)__shfl_xor((int)part, off, 32);

    // routing = part (terms < 2^10, so int64 / &MASK32 semantics collapse to u32).
    // (routing^seed)%64 for seed 0..3 enumerates {base..base+3}, base = routing & ~3 & 63,
    // so the two smallest sorted experts are always base, base+1 (distinct).
    unsigned base = part & 60u;            // 63 & ~3
    float s0 = (float)base;
    float s1 = (float)(base + 1u);

    if (lane == 0) {
        *(float2*)(sel_out + (size_t)row * 2) = make_float2(s0, s1);
        *(float2*)(w_out   + (size_t)row * 2) = make_float2(0.5f, 0.5f);
    }

    // masks: lane writes experts 2*lane, 2*lane+1 (coalesced 256B per wave)
    unsigned e0 = (unsigned)(lane * 2), e1 = e0 + 1u;
    float2 m;
    m.x = (e0 == base || e0 == base + 1u) ? 0.5f : 0.0f;
    m.y = (e1 == base || e1 == base + 1u) ? 0.5f : 0.0f;
    *(float2*)(mask_out + (size_t)row * DLIM + (size_t)(lane * 2)) = m;
}

extern "C" void kernel_launch(void* const* d_in, const int* in_sizes, int n_in,
                              void* d_out, int out_size, void* d_ws, size_t ws_size,
                              hipStream_t stream) {
    (void)n_in; (void)out_size; (void)d_ws; (void)ws_size;
    const float* hs = (const float*)d_in[0];
    const int nrows = in_sizes[0] / DIM;   // 16384

    float* out   = (float*)d_out;
    float* sel   = out;                         // [n,2]
    float* w     = out + (size_t)nrows * 2;     // [n,2]
    float* masks = out + (size_t)nrows * 4;     // [n,64]

    const int blocks = (nrows + WAVES_PER_BLOCK - 1) / WAVES_PER_BLOCK;
    router_kernel<<<blocks, 256, 0, stream>>>(hs, sel, w, masks, nrows);
}